// PersistenceLandscapeLoss_55362128445731
// MI455X (gfx1250) — compile-verified
//
#include <hip/hip_runtime.h>
#include <hip/hip_bf16.h>

typedef __attribute__((ext_vector_type(2))) float v2f;
typedef __attribute__((ext_vector_type(8))) float v8f;

#define NPTS   4096
#define KDIM   512
#define TILE   128
#define KC     32
#define LDST   36      // padded LDS row stride (floats): rows*36 % 64 all-distinct banks
#define NT     24
#define INVT   (1.0f / 0.15f)

// ---------------------------------------------------------------- init
__global__ __launch_bounds__(64) void topo_init_kernel(unsigned int* minb,
                                                       unsigned int* maxb,
                                                       float* iso, float* edge) {
    int t = threadIdx.x;
    if (t == 0) *minb = 0x7F7FFFFFu;   // +FLT_MAX bits (dist >= 0 so bit order == float order)
    if (t == 1) *maxb = 0u;
    if (t < NT) { iso[t] = 0.0f; edge[t] = 0.0f; }
}

// ---------------------------------------------------------------- row norms
__global__ __launch_bounds__(128) void topo_rownorm_kernel(const float* __restrict__ E,
                                                           float* __restrict__ sq) {
    int row = blockIdx.x;
    int t = threadIdx.x;
    const float4 v = *(const float4*)&E[(size_t)row * KDIM + t * 4];
    float s = v.x * v.x + v.y * v.y + v.z * v.z + v.w * v.w;
#pragma unroll
    for (int off = 16; off >= 1; off >>= 1) s += __shfl_xor(s, off, 32);
    __shared__ float ws[4];
    if ((t & 31) == 0) ws[t >> 5] = s;
    __syncthreads();
    if (t == 0) sq[row] = ws[0] + ws[1] + ws[2] + ws[3];
}

// ---------------------------------------------------------------- Gram GEMM (fp32 WMMA) + distance epilogue
__global__ __launch_bounds__(256) void topo_gram_dist_kernel(const float* __restrict__ E,
                                                             const float* __restrict__ sq,
                                                             float* __restrict__ dist,
                                                             unsigned int* minb,
                                                             unsigned int* maxb) {
    __shared__ float Al[TILE * LDST];
    __shared__ float Bl[TILE * LDST];

    const int tid  = threadIdx.x;
    const int lane = tid & 31;
    const int w    = tid >> 5;        // wave 0..7
    const int l15  = lane & 15;
    const int lhi  = lane >> 4;       // 0/1: K/M half selector per WMMA layout

    const int m0 = blockIdx.x * TILE;
    const int n0 = blockIdx.y * TILE;
    const int sm = (w >> 2) * 64;     // wave subtile origin: 2x4 waves -> 64x32 each
    const int sn = (w & 3) * 32;

    v8f acc[4][2];
#pragma unroll
    for (int mi = 0; mi < 4; ++mi)
#pragma unroll
        for (int ni = 0; ni < 2; ++ni)
            acc[mi][ni] = (v8f){0.f, 0.f, 0.f, 0.f, 0.f, 0.f, 0.f, 0.f};

    for (int k0 = 0; k0 < KDIM; k0 += KC) {
        // stage A (rows m0..) and B (rows n0.. of E, used as columns) tiles: 128x32 fp32 each
#pragma unroll
        for (int i = 0; i < 4; ++i) {
            int idx = tid + i * 256;          // 0..1023 float4 slots
            int r   = idx >> 3;               // 0..127
            int c4  = (idx & 7) * 4;          // 0..28
            float4 a = *(const float4*)&E[(size_t)(m0 + r) * KDIM + k0 + c4];
            *(float4*)&Al[r * LDST + c4] = a;
            float4 b = *(const float4*)&E[(size_t)(n0 + r) * KDIM + k0 + c4];
            *(float4*)&Bl[r * LDST + c4] = b;
        }
        if (k0 + KC < KDIM)  // pull next chunk toward L2 (global_prefetch_b8)
            __builtin_prefetch(&E[(size_t)(m0 + (tid >> 1)) * KDIM + k0 + KC], 0, 1);
        __syncthreads();

#pragma unroll
        for (int kk = 0; kk < KC; kk += 4) {
            const int koff = kk + 2 * lhi;
            v2f af[4], bf[2];
#pragma unroll
            for (int mi = 0; mi < 4; ++mi)
                af[mi] = *(const v2f*)&Al[(sm + 16 * mi + l15) * LDST + koff];
#pragma unroll
            for (int ni = 0; ni < 2; ++ni)
                bf[ni] = *(const v2f*)&Bl[(sn + 16 * ni + l15) * LDST + koff];
#pragma unroll
            for (int mi = 0; mi < 4; ++mi)
#pragma unroll
                for (int ni = 0; ni < 2; ++ni)
                    acc[mi][ni] = __builtin_amdgcn_wmma_f32_16x16x4_f32(
                        false, af[mi], false, bf[ni], (short)0, acc[mi][ni], false, false);
        }
        __syncthreads();
    }

    // epilogue: dist = sqrt(relu(sq_i + sq_j - 2G)), diag -> 1e9, track off-diag min/max
    float lmin = 3.0e38f, lmax = 0.0f;
#pragma unroll
    for (int mi = 0; mi < 4; ++mi) {
#pragma unroll
        for (int ni = 0; ni < 2; ++ni) {
#pragma unroll
            for (int j = 0; j < 8; ++j) {
                int grow = m0 + sm + 16 * mi + 8 * lhi + j;
                int gcol = n0 + sn + 16 * ni + l15;
                float g  = acc[mi][ni][j];
                float d2 = sq[grow] + sq[gcol] - 2.0f * g;
                d2 = fmaxf(d2, 0.0f);
                float d = (d2 > 1e-12f) ? sqrtf(d2) : 0.0f;
                if (grow == gcol) {
                    d = 1.0e9f;
                } else {
                    lmin = fminf(lmin, d);
                    lmax = fmaxf(lmax, d);
                }
                dist[(size_t)grow * NPTS + gcol] = d;
            }
        }
    }
#pragma unroll
    for (int off = 16; off >= 1; off >>= 1) {
        lmin = fminf(lmin, __shfl_xor(lmin, off, 32));
        lmax = fmaxf(lmax, __shfl_xor(lmax, off, 32));
    }
    if (lane == 0) {
        atomicMin(minb, __float_as_uint(lmin));
        atomicMax(maxb, __float_as_uint(lmax));
    }
}

// ---------------------------------------------------------------- 24-threshold filtration pass
__global__ __launch_bounds__(256) void topo_filtration_kernel(const float* __restrict__ dist,
                                                              const unsigned int* __restrict__ minb,
                                                              const unsigned int* __restrict__ maxb,
                                                              float* iso, float* edge) {
    const int row = blockIdx.x;
    const int tid = threadIdx.x;
    const int lane = tid & 31;
    const int w = tid >> 5;

    const float dmin  = __uint_as_float(*minb);
    const float dmaxr = __uint_as_float(*maxb);
    const float dmax  = fmaxf(dmaxr, dmin + 1e-4f);
    const float stepx = ((dmax - dmin) / (float)(NT - 1)) * INVT;

    float s[NT];
#pragma unroll
    for (int t = 0; t < NT; ++t) s[t] = 0.0f;

    for (int c = tid; c < NPTS; c += 256) {
        float d = dist[(size_t)row * NPTS + c];
        if (d < 1e8f) {                       // off-diagonal only (diag = 1e9)
            float x = (dmin - d) * INVT;      // x_t = (thresh_t - d)/T, thresh_t linear
#pragma unroll
            for (int t = 0; t < NT; ++t) {
                s[t] += 1.0f / (1.0f + __expf(-x));
                x += stepx;
            }
        }
    }
#pragma unroll
    for (int t = 0; t < NT; ++t)
#pragma unroll
        for (int off = 16; off >= 1; off >>= 1)
            s[t] += __shfl_xor(s[t], off, 32);

    __shared__ float red[8][NT];
    if (lane == 0) {
#pragma unroll
        for (int t = 0; t < NT; ++t) red[w][t] = s[t];
    }
    __syncthreads();
    if (tid < NT) {
        float tot = 0.0f;
#pragma unroll
        for (int w2 = 0; w2 < 8; ++w2) tot += red[w2][tid];
        float deg = fmaxf(tot, 1e-6f);                 // clip(sum, 1e-6)
        atomicAdd(&iso[tid], (deg < 0.5f) ? 1.0f : 0.0f);
        atomicAdd(&edge[tid], tot);                    // row's adjacency mass
    }
}

// ---------------------------------------------------------------- finalize scalar
__global__ __launch_bounds__(32) void topo_finalize_kernel(const float* iso, const float* edge,
                                                           float* out) {
    if (threadIdx.x == 0) {
        float h0 = 0.0f;
        for (int t = NT - 8; t < NT; ++t) h0 += iso[t];
        h0 *= (1.0f / 8.0f);
        float h1 = 0.0f;
        for (int t = 0; t < NT; ++t) {
            float n_edges = edge[t] * 0.5f;
            h1 += fmaxf(n_edges - (float)(NPTS - 1), 0.0f) / (float)NPTS;
        }
        h1 *= (1.0f / (float)NT);
        out[0] = (h0 + 0.5f * h1) * 0.1f;
    }
}

// ---------------------------------------------------------------- launch
extern "C" void kernel_launch(void* const* d_in, const int* in_sizes, int n_in,
                              void* d_out, int out_size, void* d_ws, size_t ws_size,
                              hipStream_t stream) {
    const float* E = (const float*)d_in[0];      // 4096 x 512 fp32
    float* out = (float*)d_out;                  // scalar fp32

    // workspace layout
    float* dist = (float*)d_ws;                                  // N*N floats (64 MB)
    float* sq   = dist + (size_t)NPTS * NPTS;                    // N floats
    unsigned int* minb = (unsigned int*)(sq + NPTS);
    unsigned int* maxb = minb + 1;
    float* iso  = (float*)(maxb + 1);                            // NT floats
    float* edge = iso + NT;                                      // NT floats

    topo_init_kernel<<<1, 64, 0, stream>>>(minb, maxb, iso, edge);
    topo_rownorm_kernel<<<NPTS, 128, 0, stream>>>(E, sq);
    dim3 ggrid(NPTS / TILE, NPTS / TILE);
    topo_gram_dist_kernel<<<ggrid, 256, 0, stream>>>(E, sq, dist, minb, maxb);
    topo_filtration_kernel<<<NPTS, 256, 0, stream>>>(dist, minb, maxb, iso, edge);
    topo_finalize_kernel<<<1, 32, 0, stream>>>(iso, edge, out);
}